// OutlierInjection_21784074125438
// MI455X (gfx1250) — compile-verified
//
#include <hip/hip_runtime.h>

// ---------------------------------------------------------------------------
// Problem constants (from reference setup_inputs)
// ---------------------------------------------------------------------------
#define N_   32
#define C_   3
#define H_   512
#define W_   512
#define HW_  (H_ * W_)           // 262144 pixels per channel
#define NC_  (N_ * C_)           // 96 channels
#define PIX_TOTAL (NC_ * HW_)    // 25165824
#define IGNORE_LABEL_ 100

typedef __attribute__((ext_vector_type(2))) float v2f;
typedef __attribute__((ext_vector_type(8))) float v8f;

#if defined(__has_builtin)
# if __has_builtin(__builtin_amdgcn_wmma_f32_16x16x4_f32)
#  define HAVE_WMMA4 1
# endif
#endif

// ---------------------------------------------------------------------------
// CDNA5 async global->LDS copy (ASYNCcnt-tracked). Each lane copies 16 bytes.
// LDS byte offset = low 32 bits of the generic pointer (shared aperture).
// ---------------------------------------------------------------------------
__device__ __forceinline__ void async_copy_b128(const int* gaddr, int* lds_ptr) {
    unsigned lds_off = (unsigned)(unsigned long long)lds_ptr;
    asm volatile("global_load_async_to_lds_b128 %0, %1, off"
                 :: "v"(lds_off), "v"(gaddr)
                 : "memory");
}

// ---------------------------------------------------------------------------
// Kernel 0: zero the histogram workspace (d_ws is poisoned by the harness)
// ---------------------------------------------------------------------------
__global__ void zero_hist_kernel(int* __restrict__ p, int n) {
    int i = blockIdx.x * blockDim.x + threadIdx.x;
    if (i < n) p[i] = 0;
}

// ---------------------------------------------------------------------------
// Kernel 1: per-channel 256-bin histograms of outlier (src) and image (tgt).
// Per-wave double-buffered async staging into LDS; LDS-privatized histograms
// merged to global with atomics.
// ---------------------------------------------------------------------------
#define BPC   16                  // blocks per channel
#define TPB   256                 // 8 waves of 32
#define PPB   (HW_ / BPC)         // 16384 pixels per block
#define PPI   (TPB * 4)           // 1024 pixels per iteration (b128/thread)
#define ITERS (PPB / PPI)         // 16

__global__ void __launch_bounds__(TPB)
hist_kernel(const int* __restrict__ image, const int* __restrict__ outlier,
            int* __restrict__ g_hs, int* __restrict__ g_ht) {
    __shared__ int s_hs[256];
    __shared__ int s_ht[256];
    __shared__ __align__(16) int stage[2][2][PPI];   // [buf][src/tgt][pixels]

    const int t   = threadIdx.x;
    const int ch  = blockIdx.x / BPC;
    const int blk = blockIdx.x % BPC;
    const int base = ch * HW_ + blk * PPB;

    s_hs[t] = 0;
    s_ht[t] = 0;
    __syncthreads();

    // Prologue: stage iteration 0 (2 async ops per wave).
    async_copy_b128(outlier + base + t * 4, &stage[0][0][t * 4]);
    async_copy_b128(image   + base + t * 4, &stage[0][1][t * 4]);

    for (int i = 0; i < ITERS; ++i) {
        if (i + 1 < ITERS) {
            // Stage next tile while binning current one.
            const int off = base + (i + 1) * PPI + t * 4;
            async_copy_b128(outlier + off, &stage[(i + 1) & 1][0][t * 4]);
            async_copy_b128(image   + off, &stage[(i + 1) & 1][1][t * 4]);
            // 2 ops of iter i+1 outstanding -> iter i's data has landed.
            asm volatile("s_wait_asynccnt 0x2" ::: "memory");
        } else {
            asm volatile("s_wait_asynccnt 0x0" ::: "memory");
        }
        const int b = i & 1;
        #pragma unroll
        for (int j = 0; j < 4; ++j) {
            const int sv = stage[b][0][t * 4 + j] & 255;
            const int tv = stage[b][1][t * 4 + j] & 255;
            atomicAdd(&s_hs[sv], 1);     // ds_add_u32
            atomicAdd(&s_ht[tv], 1);
        }
    }
    __syncthreads();
    atomicAdd(&g_hs[ch * 256 + t], s_hs[t]);
    atomicAdd(&g_ht[ch * 256 + t], s_ht[t]);
}

// ---------------------------------------------------------------------------
// 256-element inclusive cumsum via WMMA:
// view h as 16x16 matrix A (A[i][j] = h[16i+j]); A x L (L upper-triangular
// ones) done as 4 chained v_wmma_f32_16x16x4_f32, then add row offsets.
// Counts <= 2^18 are exact in f32.
// ---------------------------------------------------------------------------
__device__ void cumsum256(const float* vals, float* out, float* rowoff, int t) {
    __syncthreads();
#if defined(HAVE_WMMA4)
    if (t < 32) {                         // wave 0, EXEC all ones
        const int ln15 = t & 15;          // A row (M) and B column (N)
        const int koff = (t >= 16) ? 2 : 0;
        v8f acc = {};
        #pragma unroll
        for (int k = 0; k < 4; ++k) {
            const int r0 = 4 * k + koff;
            v2f a, b;
            a.x = vals[ln15 * 16 + r0];           // A[M][r0]
            a.y = vals[ln15 * 16 + r0 + 1];       // A[M][r0+1]
            b.x = (r0     <= ln15) ? 1.0f : 0.0f; // L[r0][N]
            b.y = (r0 + 1 <= ln15) ? 1.0f : 0.0f; // L[r0+1][N]
            acc = __builtin_amdgcn_wmma_f32_16x16x4_f32(
                false, a, false, b, (short)0, acc, false, false);
        }
        #pragma unroll
        for (int v = 0; v < 8; ++v) {
            const int row = v + ((t >= 16) ? 8 : 0);
            out[row * 16 + ln15] = acc[v];        // row-wise prefix sums
        }
    }
    __syncthreads();
    if (t == 0) {                          // exclusive scan of row totals
        float o = 0.0f;
        for (int r = 0; r < 16; ++r) { rowoff[r] = o; o += out[r * 16 + 15]; }
    }
    __syncthreads();
    out[t] += rowoff[t >> 4];
    __syncthreads();
#else
    (void)rowoff;
    if (t == 0) {
        float s = 0.0f;
        for (int v = 0; v < 256; ++v) { s += vals[v]; out[v] = s; }
    }
    __syncthreads();
#endif
}

// ---------------------------------------------------------------------------
// Kernel 2: per-channel histogram-matching LUT, replicating _match_lut.
// One 256-thread block per channel (96 blocks) -- negligible runtime.
// ---------------------------------------------------------------------------
__global__ void __launch_bounds__(256)
lut_kernel(const int* __restrict__ g_hs, const int* __restrict__ g_ht,
           int* __restrict__ g_lut) {
    __shared__ float vals[256];
    __shared__ float cumS[256];      // becomes cpf_s numerator
    __shared__ float cumT[256];      // becomes cpf_t
    __shared__ float rowoff[16];
    __shared__ int   sht[256];
    __shared__ int   prevA[256];
    __shared__ int   nxtA[256];
    __shared__ int   zsh;

    const int ch = blockIdx.x;
    const int t  = threadIdx.x;
    const int hs = g_hs[ch * 256 + t];
    const int ht = g_ht[ch * 256 + t];
    sht[t] = ht;
    if (t == 0) zsh = hs;                       // zero_count = h_s[0]
    vals[t] = (t == 0) ? 0.0f : (float)hs;      // source_counts[0] -= z

    cumsum256(vals, cumS, rowoff, t);           // cumsum(h_s adjusted)

    vals[t] = (float)ht;
    cumsum256(vals, cumT, rowoff, t);           // cumsum(h_t)

    const float denomS = fmaxf((float)(HW_ - zsh), 1.0f);
    const float q = cumS[t] / denomS;           // cpf_s[t]
    cumT[t] = cumT[t] / (float)HW_;             // cpf_t[t]

    if (t == 0) {                               // prev: last present idx <= v
        int p = -1;
        for (int v = 0; v < 256; ++v) { if (sht[v] > 0) p = v; prevA[v] = p; }
    } else if (t == 32) {                       // nxt: first present idx >= v
        int nx = 256;
        for (int v = 255; v >= 0; --v) { if (sht[v] > 0) nx = v; nxtA[v] = nx; }
    }
    __syncthreads();

    // searchsorted(cpf_t, q, side='left'): first i with cpf_t[i] >= q
    int lo = 0, hi = 256;
    while (lo < hi) {
        const int mid = (lo + hi) >> 1;
        if (cumT[mid] < q) lo = mid + 1; else hi = mid;
    }
    const int  iu_raw = lo;
    const bool over   = (iu_raw >= 256);
    const int  iu     = nxtA[iu_raw > 255 ? 255 : iu_raw];
    const int  iu_c   = iu > 255 ? 255 : iu;
    const int  vlo    = (iu > 0) ? prevA[(iu - 1) > 255 ? 255 : (iu - 1)] : -1;
    const float x_lo  = cumT[vlo < 0 ? 0 : vlo];
    const float x_hi  = cumT[iu_c];
    const float fi    = (float)iu;
    const float fv    = (float)vlo;
    const float den   = fmaxf(x_hi - x_lo, 1e-12f);
    float f = (vlo < 0) ? fi : (fv + (q - x_lo) * (fi - fv) / den);
    if (over) f = (float)prevA[255];
    g_lut[ch * 256 + t] = (int)f;               // torch-style truncation
}

// ---------------------------------------------------------------------------
// Kernel 3: apply blend + label overwrite. b128-vectorized, 4 px/thread.
// ---------------------------------------------------------------------------
__device__ __forceinline__ float apply1(int iv, int ov, float m, const int* lc) {
    const float fi = (float)iv;
    const float mt = (float)lc[ov & 255];
    const float f  = fi - m * fi + mt;          // ALPHA = 1
    const int   ii = (int)f;                    // trunc toward zero
    return (float)(ii & 255);                   // uint8 wrap -> int32 -> f32
}

__global__ void __launch_bounds__(256)
apply_kernel(const int* __restrict__ image, const int* __restrict__ label,
             const int* __restrict__ outlier, const float* __restrict__ mask,
             const int* __restrict__ lut,
             float* __restrict__ out_img, float* __restrict__ out_label) {
    const int tid = blockIdx.x * blockDim.x + threadIdx.x;
    const int p   = tid * 4;                    // pixel index in [0, N*HW)
    const int n   = p / HW_;
    const int rem = p - n * HW_;

    const float4 m  = *(const float4*)(mask + p);
    const int4   lb = *(const int4*)(label + p);
    float4 lo;
    lo.x = (m.x != 0.0f) ? (float)IGNORE_LABEL_ : (float)lb.x;
    lo.y = (m.y != 0.0f) ? (float)IGNORE_LABEL_ : (float)lb.y;
    lo.z = (m.z != 0.0f) ? (float)IGNORE_LABEL_ : (float)lb.z;
    lo.w = (m.w != 0.0f) ? (float)IGNORE_LABEL_ : (float)lb.w;
    *(float4*)(out_label + p) = lo;

    #pragma unroll
    for (int c = 0; c < 3; ++c) {
        const int base = (n * 3 + c) * HW_ + rem;
        const int4 iv = *(const int4*)(image + base);
        const int4 ov = *(const int4*)(outlier + base);
        const int* lc = lut + (n * 3 + c) * 256;   // L2-resident
        float4 r;
        r.x = apply1(iv.x, ov.x, m.x, lc);
        r.y = apply1(iv.y, ov.y, m.y, lc);
        r.z = apply1(iv.z, ov.z, m.z, lc);
        r.w = apply1(iv.w, ov.w, m.w, lc);
        *(float4*)(out_img + base) = r;
    }
}

// ---------------------------------------------------------------------------
// Host launcher
// ---------------------------------------------------------------------------
extern "C" void kernel_launch(void* const* d_in, const int* in_sizes, int n_in,
                              void* d_out, int out_size, void* d_ws, size_t ws_size,
                              hipStream_t stream) {
    (void)in_sizes; (void)n_in; (void)out_size; (void)ws_size;

    const int*   image   = (const int*)d_in[0];
    const int*   label   = (const int*)d_in[1];
    const int*   outlier = (const int*)d_in[2];
    const float* mask    = (const float*)d_in[3];

    int* hs  = (int*)d_ws;            // [96][256] source (outlier) histograms
    int* ht  = hs + NC_ * 256;        // [96][256] target (image) histograms
    int* lut = ht + NC_ * 256;        // [96][256] match LUTs

    float* out_img = (float*)d_out;              // [32,3,512,512]
    float* out_lab = out_img + PIX_TOTAL;        // [32,512,512]

    const int nz = NC_ * 256 * 2;
    zero_hist_kernel<<<(nz + 255) / 256, 256, 0, stream>>>(hs, nz);
    hist_kernel<<<NC_ * BPC, TPB, 0, stream>>>(image, outlier, hs, ht);
    lut_kernel<<<NC_, 256, 0, stream>>>(hs, ht, lut);
    apply_kernel<<<(N_ * HW_ / 4) / 256, 256, 0, stream>>>(
        image, label, outlier, mask, lut, out_img, out_lab);
}